// CNNRandomProjection_81887846465685
// MI455X (gfx1250) — compile-verified
//
#include <hip/hip_runtime.h>
#include <hip/hip_bf16.h>

typedef __attribute__((ext_vector_type(2))) float v2f;
typedef __attribute__((ext_vector_type(8))) float v8f;

#define C_CH   128
#define BN_N   (256.0f * 32.0f * 32.0f)   // B*H*W elements per channel

// ---------------------------------------------------------------------------
// Pass 1: per-(b,c) 32x32x32 GEMM via V_WMMA_F32_16X16X4_F32, scale + leaky,
// write Y, accumulate per-channel sum / sumsq.
// One wave (32 lanes) per (b,c) tile; 8 waves per block.
// ---------------------------------------------------------------------------
__global__ __launch_bounds__(256) void proj_pass1(
    const float* __restrict__ x,     // [B,C,32,32]
    const float* __restrict__ P,     // [C,32,32]
    const float* __restrict__ lam,   // [1]
    float* __restrict__ y,           // [B,C,32,32] (d_out)
    float* __restrict__ stats)       // ws: [0,128)=sum, [128,256)=sumsq
{
    const int wave = threadIdx.x >> 5;
    const int lane = threadIdx.x & 31;
    const int tile = blockIdx.x * 8 + wave;      // flat (b*C + c)
    const int c    = tile & (C_CH - 1);

    const float* __restrict__ Xp = x + (size_t)tile * 1024;  // 32x32 row-major (h,w)
    const float* __restrict__ Pp = P + (size_t)c    * 1024;  // 32x32 row-major (k,h)
    float* __restrict__ Yp = y + (size_t)tile * 1024;

    const int half = lane >> 4;     // 0: K=0..1, 1: K=2..3 within a 4-wide K step
    const int l16  = lane & 15;

    v8f acc00 = {}, acc01 = {}, acc10 = {}, acc11 = {};

#pragma unroll
    for (int h0 = 0; h0 < 32; h0 += 4) {
        const int kb = h0 + half * 2;
        // A fragments (P tile): A[m][k] = Pp[m*32 + k]; two contiguous floats.
        const v2f a0 = *reinterpret_cast<const v2f*>(Pp + (l16      ) * 32 + kb);
        const v2f a1 = *reinterpret_cast<const v2f*>(Pp + (l16 + 16 ) * 32 + kb);
        // B fragments (X tile): B[k][n] = Xp[k*32 + n]; rows K=kb, kb+1.
        v2f b0, b1;
        b0.x = Xp[(kb    ) * 32 + l16     ];
        b0.y = Xp[(kb + 1) * 32 + l16     ];
        b1.x = Xp[(kb    ) * 32 + l16 + 16];
        b1.y = Xp[(kb + 1) * 32 + l16 + 16];

        acc00 = __builtin_amdgcn_wmma_f32_16x16x4_f32(false, a0, false, b0, (short)0, acc00, false, false);
        acc01 = __builtin_amdgcn_wmma_f32_16x16x4_f32(false, a0, false, b1, (short)0, acc01, false, false);
        acc10 = __builtin_amdgcn_wmma_f32_16x16x4_f32(false, a1, false, b0, (short)0, acc10, false, false);
        acc11 = __builtin_amdgcn_wmma_f32_16x16x4_f32(false, a1, false, b1, (short)0, acc11, false, false);
    }

    const float s = lam[0];  // * SQRT_D (== 1.0)
    float sum = 0.0f, sq = 0.0f;

    // C/D layout: VGPR r, lanes 0-15 -> M = mt*16 + r, lanes 16-31 -> M = mt*16 + 8 + r.
    const int rowBase = half * 8;
    const int col0 = l16;

#pragma unroll
    for (int r = 0; r < 8; ++r) {
        float t;
        t = s * acc00[r]; t = (t >= 0.0f) ? t : 0.2f * t;
        Yp[(rowBase + r) * 32 + col0] = t;            sum += t; sq += t * t;
        t = s * acc01[r]; t = (t >= 0.0f) ? t : 0.2f * t;
        Yp[(rowBase + r) * 32 + col0 + 16] = t;       sum += t; sq += t * t;
        t = s * acc10[r]; t = (t >= 0.0f) ? t : 0.2f * t;
        Yp[(16 + rowBase + r) * 32 + col0] = t;       sum += t; sq += t * t;
        t = s * acc11[r]; t = (t >= 0.0f) ? t : 0.2f * t;
        Yp[(16 + rowBase + r) * 32 + col0 + 16] = t;  sum += t; sq += t * t;
    }

    // wave32 butterfly reduction
#pragma unroll
    for (int o = 16; o > 0; o >>= 1) {
        sum += __shfl_xor(sum, o, 32);
        sq  += __shfl_xor(sq,  o, 32);
    }
    if (lane == 0) {
        atomicAdd(&stats[c], sum);
        atomicAdd(&stats[C_CH + c], sq);
    }
}

// ---------------------------------------------------------------------------
// Pass 2: fold channel stats into affine A*y + B.
// ---------------------------------------------------------------------------
__global__ void bn_stats(const float* __restrict__ stats,
                         const float* __restrict__ gamma,
                         const float* __restrict__ beta,
                         float* __restrict__ ab)   // ws: [0,128)=A, [128,256)=B
{
    const int cc = threadIdx.x;     // 128 threads
    const float mean = stats[cc] * (1.0f / BN_N);
    const float var  = stats[C_CH + cc] * (1.0f / BN_N) - mean * mean;
    const float inv  = rsqrtf(var + 1e-5f);
    const float A    = gamma[cc] * inv;
    ab[cc]        = A;
    ab[C_CH + cc] = beta[cc] - A * mean;
}

// ---------------------------------------------------------------------------
// Pass 3: y = A[c]*y + B[c], float4 vectorized (1024 elems per (b,c) plane,
// so all 4 lanes of a float4 share one channel).
// ---------------------------------------------------------------------------
__global__ __launch_bounds__(256) void bn_apply(float* __restrict__ y,
                                                const float* __restrict__ ab)
{
    const int i  = blockIdx.x * blockDim.x + threadIdx.x;   // float4 index
    const int cc = ((i << 2) >> 10) & (C_CH - 1);
    const float A = ab[cc];
    const float B = ab[C_CH + cc];
    float4 v = reinterpret_cast<float4*>(y)[i];
    v.x = A * v.x + B;
    v.y = A * v.y + B;
    v.z = A * v.z + B;
    v.w = A * v.w + B;
    reinterpret_cast<float4*>(y)[i] = v;
}

extern "C" void kernel_launch(void* const* d_in, const int* in_sizes, int n_in,
                              void* d_out, int out_size, void* d_ws, size_t ws_size,
                              hipStream_t stream)
{
    const float* x     = (const float*)d_in[0];
    const float* P     = (const float*)d_in[1];
    const float* lam   = (const float*)d_in[2];
    const float* gamma = (const float*)d_in[3];
    const float* beta  = (const float*)d_in[4];
    float* y = (float*)d_out;

    float* stats = (float*)d_ws;            // 256 floats: sum | sumsq
    float* ab    = stats + 2 * C_CH;        // 256 floats: A | B

    // Zero the per-channel accumulators (graph-capturable).
    hipMemsetAsync(stats, 0, 2 * C_CH * sizeof(float), stream);

    // B*C = 32768 (b,c) tiles, 8 waves per block -> 4096 blocks.
    proj_pass1<<<4096, 256, 0, stream>>>(x, P, lam, y, stats);
    bn_stats<<<1, C_CH, 0, stream>>>(stats, gamma, beta, ab);
    // 33,554,432 elements / 4 = 8,388,608 float4 -> 32768 blocks of 256.
    bn_apply<<<32768, 256, 0, stream>>>(y, ab);
}